// EdgeConv_4183298146831
// MI455X (gfx1250) — compile-verified
//
#include <hip/hip_runtime.h>
#include <math.h>

#define BB 4
#define CC 64
#define NN 4096
#define OO 64
#define KK 20
#define EPSV 1e-5f

typedef __attribute__((ext_vector_type(2))) float v2f;
typedef __attribute__((ext_vector_type(8))) float v8f;

// D = A(16x4 f32) * B(4x16 f32) + C(16x16 f32), wave32
__device__ __forceinline__ v8f wmma_f32(v2f a, v2f b, v8f c) {
    return __builtin_amdgcn_wmma_f32_16x16x4_f32(false, a, false, b, (short)0, c,
                                                 false, false);
}

// ---------------- prep: x (B,C,N) -> xt (B,N,C) and xx[b][n] = sum_c x^2 ----
__global__ __launch_bounds__(256) void prep_kernel(const float* __restrict__ x,
                                                   float* __restrict__ xt,
                                                   float* __restrict__ xx) {
    int t = blockIdx.x * blockDim.x + threadIdx.x;  // 0 .. B*N-1
    int b = t / NN;
    int n = t - b * NN;
    const float* px = x + (size_t)b * CC * NN + n;  // stride NN over c
    float* pt = xt + (size_t)t * CC;
    float s = 0.f;
#pragma unroll 8
    for (int c = 0; c < CC; ++c) {
        float v = px[(size_t)c * NN];
        pt[c] = v;
        s += v * v;
    }
    xx[t] = s;
}

// ---------------- kNN: per wave, 32 query rows; top-20 by -||xi-xj||^2 -------
__global__ __launch_bounds__(128) void knn_kernel(const float* __restrict__ xt,
                                                  const float* __restrict__ xx,
                                                  int* __restrict__ nbr) {
    const int wave = threadIdx.x >> 5;
    const int lane = threadIdx.x & 31;
    const int laneM = lane & 15;
    const int kofs = (lane < 16) ? 0 : 2;

    const int row0 = blockIdx.x * 128 + wave * 32;  // 128 rows/block, same b
    const int b = row0 / NN;
    const int r0 = row0 - b * NN;

    // Preload A fragments for both 16-row tiles (invariant over column tiles).
    const float* prowLo = xt + (size_t)(b * NN + r0 + laneM) * CC;
    const float* prowHi = prowLo + 16 * CC;
    v2f aLo[16], aHi[16];
#pragma unroll
    for (int ks = 0; ks < 16; ++ks) {
        int kc = ks * 4 + kofs;
        aLo[ks] = *(const v2f*)(prowLo + kc);
        aHi[ks] = *(const v2f*)(prowHi + kc);
    }

    float topv[KK];
    int topi[KK];
#pragma unroll
    for (int s = 0; s < KK; ++s) { topv[s] = -3.0e38f; topi[s] = 0; }
    const float xxrow = xx[b * NN + r0 + lane];  // lane owns local row `lane`

    __shared__ float dist[4][32 * 16];

    for (int ct = 0; ct < NN / 16; ++ct) {
        const int col = ct * 16 + laneM;
        const float* pcol = xt + (size_t)(b * NN + col) * CC;
        v8f accLo = {};
        v8f accHi = {};
#pragma unroll
        for (int ks = 0; ks < 16; ++ks) {
            v2f bb = *(const v2f*)(pcol + ks * 4 + kofs);
            accLo = wmma_f32(aLo[ks], bb, accLo);
            accHi = wmma_f32(aHi[ks], bb, accHi);
        }
        const float xcol = xx[b * NN + col];
#pragma unroll
        for (int j = 0; j < 8; ++j) {
            int M = j + ((lane < 16) ? 0 : 8);
            dist[wave][M * 16 + laneM] = 2.0f * accLo[j] - xcol;
            dist[wave][(16 + M) * 16 + laneM] = 2.0f * accHi[j] - xcol;
        }
        __syncthreads();
        // Lane `lane` owns query row r0+lane: scan 16 candidates.
#pragma unroll
        for (int c = 0; c < 16; ++c) {
            float v = dist[wave][lane * 16 + c] - xxrow;
            if (v > topv[KK - 1]) {  // beats current k-th best -> bubble insert
                float cv = v;
                int ci = ct * 16 + c;
#pragma unroll
                for (int s = 0; s < KK; ++s) {
                    if (cv > topv[s]) {
                        float tv = topv[s]; topv[s] = cv; cv = tv;
                        int ti = topi[s]; topi[s] = ci; ci = ti;
                    }
                }
            }
        }
        __syncthreads();
    }
    int* outp = nbr + (size_t)(b * NN + r0 + lane) * KK;
#pragma unroll
    for (int s = 0; s < KK; ++s) outp[s] = topi[s];
}

// ------ fused: gather edge -> conv1+BN+ReLU -> conv2+BN+ReLU -> max over k ---
__global__ __launch_bounds__(128) void edgeconv_kernel(
    const float* __restrict__ xt, const int* __restrict__ nbr,
    const float* __restrict__ w1, const float* __restrict__ g1,
    const float* __restrict__ be1, const float* __restrict__ mu1,
    const float* __restrict__ va1, const float* __restrict__ w2,
    const float* __restrict__ g2, const float* __restrict__ be2,
    const float* __restrict__ mu2, const float* __restrict__ va2,
    float* __restrict__ out) {
    const int wave = threadIdx.x >> 5;
    const int lane = threadIdx.x & 31;
    const int laneM = lane & 15;
    const int kofs = (lane < 16) ? 0 : 2;
    const int b = blockIdx.x >> 8;     // N/16 = 256 point-tiles per batch
    const int pt = blockIdx.x & 255;

    __shared__ float h1s[4][16 * OO];  // per-wave 16x64 intermediate
    __shared__ unsigned om[16 * OO];   // 16 points x 64 channels, fp32 bits
    for (int i = threadIdx.x; i < 16 * OO; i += 128) om[i] = 0u;
    __syncthreads();

    for (int it = 0; it < 5; ++it) {
        const int mt = wave * 5 + it;        // M-tile 0..19 (320 rows total)
        const int lrow = mt * 16 + laneM;    // this lane's A row (local)
        const int p = lrow / KK;             // point within tile 0..15
        const int kk = lrow - p * KK;        // neighbor slot 0..19
        const int n = pt * 16 + p;
        const int nb = nbr[(size_t)(b * NN + n) * KK + kk];
        const float* pc = xt + (size_t)(b * NN + n) * CC;
        const float* pn = xt + (size_t)(b * NN + nb) * CC;

        v8f acc1[4] = {};
        // edge cols 0..63: nbr - center
#pragma unroll
        for (int ks = 0; ks < 16; ++ks) {
            int kc = ks * 4 + kofs;
            v2f a = *(const v2f*)(pn + kc) - *(const v2f*)(pc + kc);
#pragma unroll
            for (int nt = 0; nt < 4; ++nt) {
                int o = nt * 16 + laneM;
                v2f bb = *(const v2f*)(w1 + o * 128 + kc);
                acc1[nt] = wmma_f32(a, bb, acc1[nt]);
            }
        }
        // edge cols 64..127: center
#pragma unroll
        for (int ks = 0; ks < 16; ++ks) {
            int kc = ks * 4 + kofs;
            v2f a = *(const v2f*)(pc + kc);
#pragma unroll
            for (int nt = 0; nt < 4; ++nt) {
                int o = nt * 16 + laneM;
                v2f bb = *(const v2f*)(w1 + o * 128 + 64 + kc);
                acc1[nt] = wmma_f32(a, bb, acc1[nt]);
            }
        }
        // BN1 (eval) + ReLU -> LDS (rows of current M-tile)
#pragma unroll
        for (int nt = 0; nt < 4; ++nt) {
            int o = nt * 16 + laneM;
            float sc = g1[o] * rsqrtf(va1[o] + EPSV);
            float sh = be1[o] - mu1[o] * sc;
#pragma unroll
            for (int j = 0; j < 8; ++j) {
                int M = j + ((lane < 16) ? 0 : 8);
                h1s[wave][M * OO + o] = fmaxf(acc1[nt][j] * sc + sh, 0.0f);
            }
        }
        __syncthreads();

        v8f acc2[4] = {};
#pragma unroll
        for (int ks = 0; ks < 16; ++ks) {
            int kc = ks * 4 + kofs;
            v2f a = *(const v2f*)(&h1s[wave][laneM * OO + kc]);
#pragma unroll
            for (int nt = 0; nt < 4; ++nt) {
                int o = nt * 16 + laneM;
                v2f bb = *(const v2f*)(w2 + o * OO + kc);
                acc2[nt] = wmma_f32(a, bb, acc2[nt]);
            }
        }
        // BN2 + ReLU + max over k (values >= 0 => uint-bit max == float max)
#pragma unroll
        for (int nt = 0; nt < 4; ++nt) {
            int o = nt * 16 + laneM;
            float sc = g2[o] * rsqrtf(va2[o] + EPSV);
            float sh = be2[o] - mu2[o] * sc;
#pragma unroll
            for (int j = 0; j < 8; ++j) {
                int M = j + ((lane < 16) ? 0 : 8);
                int lr = mt * 16 + M;
                int pp = lr / KK;
                float val = fmaxf(acc2[nt][j] * sc + sh, 0.0f);
                atomicMax(&om[pp * OO + o], __float_as_uint(val));
            }
        }
        __syncthreads();
    }
    // out[b][o][pt*16 + p]
    for (int i = threadIdx.x; i < 16 * OO; i += 128) {
        int p = i >> 6;
        int o = i & 63;
        out[(size_t)(b * OO + o) * NN + pt * 16 + p] = __uint_as_float(om[p * OO + o]);
    }
}

extern "C" void kernel_launch(void* const* d_in, const int* in_sizes, int n_in,
                              void* d_out, int out_size, void* d_ws, size_t ws_size,
                              hipStream_t stream) {
    const float* x   = (const float*)d_in[0];
    const float* w1  = (const float*)d_in[1];
    const float* g1  = (const float*)d_in[2];
    const float* be1 = (const float*)d_in[3];
    const float* mu1 = (const float*)d_in[4];
    const float* va1 = (const float*)d_in[5];
    const float* w2  = (const float*)d_in[6];
    const float* g2  = (const float*)d_in[7];
    const float* be2 = (const float*)d_in[8];
    const float* mu2 = (const float*)d_in[9];
    const float* va2 = (const float*)d_in[10];
    float* out = (float*)d_out;

    // workspace layout: xt (4 MB) | xx (64 KB) | nbr idx (1.31 MB)
    float* xt = (float*)d_ws;
    float* xx = xt + (size_t)BB * NN * CC;
    int* nbr = (int*)(xx + (size_t)BB * NN);

    prep_kernel<<<(BB * NN) / 256, 256, 0, stream>>>(x, xt, xx);
    knn_kernel<<<(BB * NN) / 128, 128, 0, stream>>>(xt, xx, nbr);
    edgeconv_kernel<<<BB * (NN / 16), 128, 0, stream>>>(
        xt, nbr, w1, g1, be1, mu1, va1, w2, g2, be2, mu2, va2, out);
}